// SPES_ResNet_32091995635793
// MI455X (gfx1250) — compile-verified
//
#include <hip/hip_runtime.h>
#include <cmath>

// ---------------- constants from the reference ----------------
#define BATCH 64
#define HDIM  100
#define WDIM  2049
#define ICH   40
#define SIGLEN 2048
// 1/sqrt(1+1e-5): BN scale factor (eps folded)
#define BN_RS 0.99999500003749973f

typedef float v2f __attribute__((ext_vector_type(2)));
typedef float v8f __attribute__((ext_vector_type(8)));

// =================================================================
// Kernel 1: masked-Gumbel top-40 per batch row (one wave32 / batch)
// =================================================================
__global__ __launch_bounds__(32)
void topk_kernel(const float* __restrict__ x, const float* __restrict__ u,
                 int* __restrict__ idx_out)
{
    const int b = blockIdx.x;
    const int lane = threadIdx.x;
    __shared__ float sc[HDIM];
    __shared__ int   sel[ICH];

    for (int h = lane; h < HDIM; h += 32) {
        // dist = x[0][b][0][h][0]
        float dist = x[((size_t)b * 2 * HDIM + h) * WDIM];
        float uu   = u[b * HDIM + h];
        float g    = -logf(-logf(uu));
        sc[h] = (dist != 0.0f) ? g : -INFINITY;
    }
    __syncthreads();

    for (int r = 0; r < ICH; ++r) {
        float best = -INFINITY; int bi = -1;
        for (int h = lane; h < HDIM; h += 32) {
            float v = sc[h];
            if (v > best) { best = v; bi = h; }
        }
        // wave32 argmax reduction (ties -> smaller index, matches top_k)
        for (int off = 16; off > 0; off >>= 1) {
            float ov = __shfl_down(best, off);
            int   oi = __shfl_down(bi,   off);
            if (ov > best || (ov == best && oi != -1 && (bi == -1 || oi < bi))) {
                best = ov; bi = oi;
            }
        }
        bi = __shfl(bi, 0);
        if (lane == 0) { sel[r] = bi; if (bi >= 0) sc[bi] = -INFINITY; }
        __syncthreads();
    }
    if (lane == 0) {
        // sort ascending (idx = jnp.sort(idx))
        for (int i = 1; i < ICH; ++i) {
            int v = sel[i], j = i - 1;
            while (j >= 0 && sel[j] > v) { sel[j + 1] = sel[j]; --j; }
            sel[j + 1] = v;
        }
        for (int i = 0; i < ICH; ++i) idx_out[b * ICH + i] = sel[i];
    }
}

// =================================================================
// Kernel 2: generic conv1d + BN (+residual) (+ReLU) via f32 WMMA.
// One wave owns NTILE 16x16 f32 tiles: 16 Cout x (16*NTILE) positions.
// K = Cin*KW advances 4/step via V_WMMA_F32_16X16X4_F32:
//   - A fragment (weights) shared by all NTILE WMMAs (1 b64 load)
//   - B addresses of tiles 1..NTILE-1 = tile0 + const -> folds into
//     the ds_load immediate offset; ci/tap decomposition once/step
//   - NTILE independent accumulator chains keep the XDL pipe busy
// Input window (Cin rows x S floats) staged in LDS once; bounds and
// stem gather resolved during staging.
// =================================================================
template<int KW, int STRIDE, int NTILE, bool GATHER>
__global__ __launch_bounds__(32)
void conv_bn_wmma(const float* __restrict__ X,   // (B,Cin,Lin) when !GATHER
                  const float* __restrict__ Xg,  // raw x base when GATHER
                  const int*   __restrict__ gidx,// (B,ICH) when GATHER
                  const float* __restrict__ W,   // (Cout,Cin,KW) == (Cout,Ktot)
                  const float* __restrict__ Gm,  // bn gamma (Cout)
                  const float* __restrict__ Bt,  // bn beta  (Cout)
                  const float* __restrict__ Res, // (B,Cout,Lout) or null
                  float* __restrict__ Y,         // (B,Cout,Lout)
                  int Cin, int Lin, int Cout, int Lout, int relu)
{
    constexpr int PAD  = KW / 2;
    constexpr int NPOS = 16 * NTILE;              // output positions per wave
    constexpr int S    = (NPOS - 1) * STRIDE + KW;// input span per Cin row
    constexpr int Spad = S | 1;                   // odd -> LDS bank spread

    extern __shared__ float smem[];               // Cin * Spad floats
    __shared__ int gcache[ICH];

    const int lane = threadIdx.x;
    const int nl   = lane & 15;
    const int half = lane >> 4;
    const int b    = blockIdx.z;
    const int m0   = blockIdx.y << 4;
    const int n0   = blockIdx.x * NPOS;
    const int Ktot = Cin * KW;
    const int base = n0 * STRIDE - PAD;           // leftmost input pos of window

    if (GATHER) {
        for (int t = lane; t < ICH; t += 32) gcache[t] = gidx[b * ICH + t];
        __syncthreads();
    }

    // ---- stage input window: bounds + gather resolved here ----
    for (int ci = 0; ci < Cin; ++ci) {
        const float* src;
        if (GATHER) {
            src = Xg + (size_t)b * (2 * HDIM * WDIM) + (size_t)gcache[ci] * WDIM + 1;
        } else {
            src = X + ((size_t)b * Cin + ci) * Lin;
        }
        for (int o = lane; o < S; o += 32) {
            const int pos = base + o;
            smem[ci * Spad + o] = (pos >= 0 && pos < Lin) ? src[pos] : 0.0f;
        }
    }
    __syncthreads();

    // ---- K loop ----
    const v2f* wrow = (const v2f*)(W + (size_t)(m0 + nl) * Ktot);  // A row (M=nl)
    const int lb = nl * STRIDE;                                    // lane B offset
    v8f acc[NTILE];
#pragma unroll
    for (int t = 0; t < NTILE; ++t) acc[t] = {};

#pragma unroll 2
    for (int kk = 0; kk < Ktot; kk += 4) {
        const int kA = kk + (half << 1);
        const v2f a = wrow[(kk >> 1) + half];     // b64: 8B-aligned pair

        // K -> (cin, tap), compile-time KW (cheap const-div)
        const int ci0 = kA / KW;
        const int t0  = kA - ci0 * KW;
        int t1 = t0 + 1, ci1 = ci0;
        if (t1 == KW) { t1 = 0; ++ci1; }
        const int o0 = ci0 * Spad + lb + t0;
        const int o1 = ci1 * Spad + lb + t1;

#pragma unroll
        for (int t = 0; t < NTILE; ++t) {
            v2f bv;
            bv.x = smem[o0 + t * (16 * STRIDE)];  // const -> ds_load imm offset
            bv.y = smem[o1 + t * (16 * STRIDE)];
            acc[t] = __builtin_amdgcn_wmma_f32_16x16x4_f32(
                         false, a, false, bv, (short)0, acc[t], false, false);
        }
    }

    // ---- epilogue: BN (+res) (+relu) ----
    const int mb = half << 3;
    const bool hasres = (Res != nullptr);
#pragma unroll
    for (int r = 0; r < 8; ++r) {
        const int m = m0 + mb + r;
        const float sc = Gm[m] * BN_RS;
        const float bt = Bt[m];
        const size_t rowo = ((size_t)b * Cout + m) * Lout + n0 + nl;
#pragma unroll
        for (int t = 0; t < NTILE; ++t) {
            float y = acc[t][r] * sc + bt;
            const size_t o = rowo + t * 16;
            if (hasres) y += Res[o];
            if (relu)   y = fmaxf(y, 0.0f);
            Y[o] = y;
        }
    }
}

// =================================================================
// Kernel 3: maxpool1d k=3 s=2 p=1 : (B,64,1024) -> (B,64,512)
// =================================================================
__global__ void maxpool_kernel(const float* __restrict__ in, float* __restrict__ out)
{
    const size_t i = (size_t)blockIdx.x * blockDim.x + threadIdx.x;
    const size_t total = (size_t)BATCH * 64 * 512;
    if (i >= total) return;
    const int o = (int)(i & 511);
    const size_t bc = i >> 9;
    const float* row = in + bc * 1024;
    const int p = 2 * o - 1;
    float m = -INFINITY;
#pragma unroll
    for (int t = 0; t < 3; ++t) {
        const int q = p + t;
        if (q >= 0 && q < 1024) m = fmaxf(m, row[q]);
    }
    out[i] = m;
}

// =================================================================
// Kernel 4: per-branch mean over L=64, concat(768), FC -> (B,1)
// =================================================================
__global__ __launch_bounds__(256)
void meanfc_kernel(const float* __restrict__ b0, const float* __restrict__ b1,
                   const float* __restrict__ b2, const float* __restrict__ fcw,
                   const float* __restrict__ fcb, float* __restrict__ out)
{
    const int b = blockIdx.x;
    const int t = threadIdx.x;  // 0..255 : channel within a branch
    const float* bs[3] = { b0, b1, b2 };
    float s = 0.0f;
    for (int br = 0; br < 3; ++br) {
        const float* row = bs[br] + ((size_t)b * 256 + t) * 64;
        float m = 0.0f;
        for (int l = 0; l < 64; ++l) m += row[l];
        s += (m * (1.0f / 64.0f)) * fcw[br * 256 + t];
    }
    __shared__ float red[256];
    red[t] = s;
    __syncthreads();
    for (int st = 128; st > 0; st >>= 1) {
        if (t < st) red[t] += red[t + st];
        __syncthreads();
    }
    if (t == 0) out[b] = red[0] + fcb[0];
}

// =================================================================
// Host-side launch helpers (NTILE=4 unless LDS > 64KB, then 2)
// =================================================================
template<int KW, int STRIDE, bool GATHER>
static inline void launch_conv(const float* X, const float* Xg, const int* gidx,
                               const float* W, const float* G, const float* B,
                               const float* Res, float* Y,
                               int Cin, int Lin, int Cout, int Lout, int relu,
                               hipStream_t stream)
{
    constexpr int P4 = ((63 * STRIDE + KW) | 1);
    constexpr int P2 = ((31 * STRIDE + KW) | 1);
    const size_t lds4 = (size_t)Cin * P4 * sizeof(float);
    if (lds4 <= 64 * 1024 && (Lout % 64) == 0) {
        conv_bn_wmma<KW, STRIDE, 4, GATHER>
            <<<dim3(Lout / 64, Cout / 16, BATCH), 32, lds4, stream>>>(
                X, Xg, gidx, W, G, B, Res, Y, Cin, Lin, Cout, Lout, relu);
    } else {
        const size_t lds2 = (size_t)Cin * P2 * sizeof(float);
        conv_bn_wmma<KW, STRIDE, 2, GATHER>
            <<<dim3(Lout / 32, Cout / 16, BATCH), 32, lds2, stream>>>(
                X, Xg, gidx, W, G, B, Res, Y, Cin, Lin, Cout, Lout, relu);
    }
}

static inline void conv_k_s2(int kw, const float* X, const float* W, const float* G,
                             const float* B, float* Y, int Cin, int Lin, int Cout,
                             int Lout, int relu, hipStream_t stream)
{
    switch (kw) {
    case 3: launch_conv<3,2,false>(X,nullptr,nullptr,W,G,B,nullptr,Y,Cin,Lin,Cout,Lout,relu,stream); break;
    case 5: launch_conv<5,2,false>(X,nullptr,nullptr,W,G,B,nullptr,Y,Cin,Lin,Cout,Lout,relu,stream); break;
    default:launch_conv<7,2,false>(X,nullptr,nullptr,W,G,B,nullptr,Y,Cin,Lin,Cout,Lout,relu,stream); break;
    }
}

static inline void conv_k_s1_res(int kw, const float* X, const float* W, const float* G,
                                 const float* B, const float* Res, float* Y,
                                 int Cin, int Lin, int Cout, int Lout, int relu,
                                 hipStream_t stream)
{
    switch (kw) {
    case 3: launch_conv<3,1,false>(X,nullptr,nullptr,W,G,B,Res,Y,Cin,Lin,Cout,Lout,relu,stream); break;
    case 5: launch_conv<5,1,false>(X,nullptr,nullptr,W,G,B,Res,Y,Cin,Lin,Cout,Lout,relu,stream); break;
    default:launch_conv<7,1,false>(X,nullptr,nullptr,W,G,B,Res,Y,Cin,Lin,Cout,Lout,relu,stream); break;
    }
}

// =================================================================
// Host launcher
// =================================================================
extern "C" void kernel_launch(void* const* d_in, const int* in_sizes, int n_in,
                              void* d_out, int out_size, void* d_ws, size_t ws_size,
                              hipStream_t stream)
{
    (void)in_sizes; (void)n_in; (void)out_size; (void)ws_size;

    const float* x      = (const float*)d_in[0];
    const float* u      = (const float*)d_in[1];
    const float* stem_w = (const float*)d_in[2];
    const float* stem_g = (const float*)d_in[3];
    const float* stem_b = (const float*)d_in[4];
    const float* fc_w   = (const float*)d_in[5];
    const float* fc_b   = (const float*)d_in[6];

    // ---- workspace layout (floats) ----
    float* wsf = (float*)d_ws;
    size_t off = 0;
    float* stemout = wsf + off; off += (size_t)BATCH * 64 * 1024;   // 4.19M f
    float* Hbuf    = wsf + off; off += (size_t)BATCH * 64 * 512;    // 2.10M f
    float* Tbuf    = wsf + off; off += (size_t)1 << 20;             // 1.05M f
    float* Rbuf    = wsf + off; off += (size_t)1 << 20;
    float* O0      = wsf + off; off += (size_t)1 << 20;
    float* O1      = wsf + off; off += (size_t)1 << 20;
    float* bout[3];
    for (int i = 0; i < 3; ++i) { bout[i] = wsf + off; off += (size_t)1 << 20; }
    int* idxb = (int*)(wsf + off);                                   // 64*40 ints

    // 1) top-40 channel indices per batch row
    topk_kernel<<<BATCH, 32, 0, stream>>>(x, u, idxb);

    // 2) stem: gather + conv(40->64,k7,s2,p3) + BN + ReLU  (WMMA f32)
    launch_conv<7, 2, true>(nullptr, x, idxb, stem_w, stem_g, stem_b, nullptr,
                            stemout, ICH, SIGLEN, 64, 1024, /*relu=*/1, stream);

    // 3) maxpool 3/2/1 -> (B,64,512)
    maxpool_kernel<<<(int)(((size_t)BATCH * 64 * 512 + 255) / 256), 256, 0, stream>>>(
        stemout, Hbuf);

    // 4) three branches of three basic blocks
    const int ks[3]    = { 3, 5, 7 };
    const int cins[3]  = { 64, 64, 128 };
    const int couts[3] = { 64, 128, 256 };
    for (int br = 0; br < 3; ++br) {
        const int kw = ks[br];
        const float* cur = Hbuf;
        int L = 512;
        for (int j = 0; j < 3; ++j) {
            const int ci = cins[j], co = couts[j], Lo = L / 2;
            const int ib = 7 + br * 27 + j * 9;
            const float* w1 = (const float*)d_in[ib + 0];
            const float* g1 = (const float*)d_in[ib + 1];
            const float* b1 = (const float*)d_in[ib + 2];
            const float* w2 = (const float*)d_in[ib + 3];
            const float* g2 = (const float*)d_in[ib + 4];
            const float* b2 = (const float*)d_in[ib + 5];
            const float* wd = (const float*)d_in[ib + 6];
            const float* gd = (const float*)d_in[ib + 7];
            const float* bd = (const float*)d_in[ib + 8];
            float* outb = (j == 2) ? bout[br] : ((j & 1) ? O1 : O0);

            // t = relu(bn1(conv(x, w1, s=2, p=kw/2)))
            conv_k_s2(kw, cur, w1, g1, b1, Tbuf, ci, L, co, Lo, 1, stream);
            // r = bnd(conv(x, wd, 1x1, s=2))
            launch_conv<1, 2, false>(cur, nullptr, nullptr, wd, gd, bd, nullptr,
                                     Rbuf, ci, L, co, Lo, 0, stream);
            // out = relu(bn2(conv(t, w2, s=1, p=kw/2)) + r)
            conv_k_s1_res(kw, Tbuf, w2, g2, b2, Rbuf, outb, co, Lo, co, Lo, 1, stream);

            cur = outb; L = Lo;
        }
    }

    // 5) global mean + concat + FC
    meanfc_kernel<<<BATCH, 256, 0, stream>>>(
        bout[0], bout[1], bout[2], fc_w, fc_b, (float*)d_out);
}